// ShotHead_20194936226245
// MI455X (gfx1250) — compile-verified
//
#include <hip/hip_runtime.h>
#include <math.h>

typedef __attribute__((ext_vector_type(2))) float v2f;
typedef __attribute__((ext_vector_type(8))) float v8f;

#define IN_DIM 64
#define HID    32

// ---------------------------------------------------------------------------
// Kernel 1: g[i] = relu(x[i,:] @ Wg1 + bg1) @ Wg2 + bg2 for all rows.
// One wave32 per 16-row tile; fp32 WMMA 16x16x4, K=64 in 16 steps, 2 N-tiles.
// A layout (16x4 f32): M = lane&15; vgpr0 holds K = 2*(lane>=16), vgpr1 K+1.
// B layout (4x16 f32): N = lane&15; vgpr0 holds K-row 2*(lane>=16), vgpr1 K+1.
// D layout (16x16 f32): acc[r] = D[M = r + 8*(lane>=16), N = lane&15].
// ---------------------------------------------------------------------------
__global__ __launch_bounds__(256) void gate_kernel(
    const float* __restrict__ x, const float* __restrict__ Wg1,
    const float* __restrict__ bg1, const float* __restrict__ Wg2,
    const float* __restrict__ bg2, float* __restrict__ g, int ntiles)
{
    __shared__ float sW[IN_DIM * HID];   // Wg1 row-major (64,32), 8KB
    __shared__ float sb1[HID];
    __shared__ float sW2[HID];

    int t = threadIdx.x;
    for (int i = t; i < IN_DIM * HID; i += 256) sW[i] = Wg1[i];
    if (t < HID) { sb1[t] = bg1[t]; sW2[t] = Wg2[t]; }
    __syncthreads();

    int wave = t >> 5;
    int lane = t & 31;
    int hl   = lane >> 4;        // which 16-lane half
    int l16  = lane & 15;
    int tile = blockIdx.x * 8 + wave;
    if (tile >= ntiles) return;  // wave-uniform: EXEC stays all-ones for WMMA

    const float* xrow = x + (size_t)(tile * 16 + l16) * IN_DIM;

    v8f acc0 = {};
    v8f acc1 = {};
    #pragma unroll
    for (int k = 0; k < 16; ++k) {
        int kb = 4 * k + 2 * hl;                 // this lane's K pair base
        v2f a = *(const v2f*)(xrow + kb);        // A: x[M, kb], x[M, kb+1]
        v2f b0, b1;
        b0.x = sW[kb * HID + l16];               // B: Wg1[kb,   N]
        b0.y = sW[(kb + 1) * HID + l16];         //    Wg1[kb+1, N]
        b1.x = sW[kb * HID + l16 + 16];          // N-tile 1 (cols 16..31)
        b1.y = sW[(kb + 1) * HID + l16 + 16];
        acc0 = __builtin_amdgcn_wmma_f32_16x16x4_f32(
                   false, a, false, b0, (short)0, acc0, false, false);
        acc1 = __builtin_amdgcn_wmma_f32_16x16x4_f32(
                   false, a, false, b1, (short)0, acc1, false, false);
    }

    // Epilogue: bias + relu + multiply by Wg2, then reduce over N (16 lanes/half).
    float w0 = sW2[l16], w1 = sW2[l16 + 16];
    float c0 = sb1[l16], c1 = sb1[l16 + 16];
    float p[8];
    #pragma unroll
    for (int r = 0; r < 8; ++r) {
        float h0 = fmaxf(acc0[r] + c0, 0.0f) * w0;
        float h1 = fmaxf(acc1[r] + c1, 0.0f) * w1;
        p[r] = h0 + h1;
    }
    #pragma unroll
    for (int msk = 1; msk <= 8; msk <<= 1) {
        #pragma unroll
        for (int r = 0; r < 8; ++r)
            p[r] += __shfl_xor(p[r], msk, 32);
    }
    if (l16 == 0) {
        float bias = bg2[0];
        #pragma unroll
        for (int r = 0; r < 8; ++r)
            g[tile * 16 + 8 * hl + r] = p[r] + bias;   // row M = r + 8*hl
    }
}

// ---------------------------------------------------------------------------
// Kernel 2: one block per segment (batch is sorted => contiguous run).
// Binary-search the run, segment softmax over g, weighted sum of x into hg,
// then fused 64->32(relu)->1 MLP.
// ---------------------------------------------------------------------------
__global__ __launch_bounds__(256) void seg_kernel(
    const float* __restrict__ x, const int* __restrict__ batch,
    const float* __restrict__ g,
    const float* __restrict__ Wm1, const float* __restrict__ bm1,
    const float* __restrict__ Wm2, const float* __restrict__ bm2,
    float* __restrict__ out, int n)
{
    __shared__ float red[256];
    __shared__ float hg[IN_DIM];

    int s = blockIdx.x;
    int t = threadIdx.x;

    // lower_bound(batch, s) and lower_bound(batch, s+1), redundantly per thread
    int lo = 0, hi = n;
    while (lo < hi) { int mid = (lo + hi) >> 1; if (batch[mid] < s) lo = mid + 1; else hi = mid; }
    int s0 = lo;
    hi = n;
    while (lo < hi) { int mid = (lo + hi) >> 1; if (batch[mid] < s + 1) lo = mid + 1; else hi = mid; }
    int s1 = lo;

    // 1) segment max of g
    float mx = -__builtin_inff();
    for (int i = s0 + t; i < s1; i += 256) mx = fmaxf(mx, g[i]);
    red[t] = mx; __syncthreads();
    for (int off = 128; off; off >>= 1) {
        if (t < off) red[t] = fmaxf(red[t], red[t + off]);
        __syncthreads();
    }
    mx = red[0]; __syncthreads();

    // 2) segment sum of exp(g - max)
    float sm = 0.0f;
    for (int i = s0 + t; i < s1; i += 256) sm += expf(g[i] - mx);
    red[t] = sm; __syncthreads();
    for (int off = 128; off; off >>= 1) {
        if (t < off) red[t] += red[t + off];
        __syncthreads();
    }
    float invS = 1.0f / red[0];   // empty segment: never used (loops below empty)
    __syncthreads();

    // 3) hg[col] = sum_i alpha_i * x[i, col]; 4 rows in flight, coalesced cols
    int col = t & 63, rg = t >> 6;
    float acc = 0.0f;
    for (int i = s0 + rg; i < s1; i += 4) {
        float a = expf(g[i] - mx) * invS;
        acc += a * x[(size_t)i * IN_DIM + col];
    }
    red[t] = acc; __syncthreads();
    if (t < IN_DIM)
        hg[t] = red[t] + red[64 + t] + red[128 + t] + red[192 + t];
    __syncthreads();

    // 4) logit = relu(hg @ Wm1 + bm1) @ Wm2 + bm2  (wave 0, 32 lanes)
    if (t < HID) {
        float h = bm1[t];
        #pragma unroll
        for (int c = 0; c < IN_DIM; ++c) h += hg[c] * Wm1[c * HID + t];
        h = fmaxf(h, 0.0f) * Wm2[t];
        #pragma unroll
        for (int msk = 16; msk; msk >>= 1) h += __shfl_xor(h, msk, 32);
        if (t == 0) out[s] = h + bm2[0];
    }
}

// ---------------------------------------------------------------------------
extern "C" void kernel_launch(void* const* d_in, const int* in_sizes, int n_in,
                              void* d_out, int out_size, void* d_ws, size_t ws_size,
                              hipStream_t stream)
{
    const float* x    = (const float*)d_in[0];
    const int*   batch= (const int*)  d_in[1];
    const float* Wg1  = (const float*)d_in[2];
    const float* bg1  = (const float*)d_in[3];
    const float* Wg2  = (const float*)d_in[4];
    const float* bg2  = (const float*)d_in[5];
    const float* Wm1  = (const float*)d_in[6];
    const float* bm1  = (const float*)d_in[7];
    const float* Wm2  = (const float*)d_in[8];
    const float* bm2  = (const float*)d_in[9];
    float* out  = (float*)d_out;
    float* gbuf = (float*)d_ws;          // N floats of scratch for gate scalars

    int n = in_sizes[0] / IN_DIM;        // 819200 (multiple of 16)
    int ntiles  = n / 16;
    int blocks1 = (ntiles + 7) / 8;      // 8 waves (tiles) per 256-thread block

    gate_kernel<<<blocks1, 256, 0, stream>>>(x, Wg1, bg1, Wg2, bg2, gbuf, ntiles);
    seg_kernel<<<out_size, 256, 0, stream>>>(x, batch, gbuf, Wm1, bm1, Wm2, bm2,
                                             out, n);
}